// RoIAlignAvg_326417515285
// MI455X (gfx1250) — compile-verified
//
#include <hip/hip_runtime.h>
#include <cstdint>

typedef unsigned int u32;
typedef u32 v4u __attribute__((ext_vector_type(4)));
typedef u32 v8u __attribute__((ext_vector_type(8)));

#define KROIS   2000
#define CCH     512
#define HF      64
#define WF      64
#define OHW     7          // output 7x7
#define SCALE_F 0.0625f
#define CHUNK   16         // channels per TDM tile
#define NCHUNK  (CCH / CHUNK)   // 32
#define TH      22         // tile rows  (roi span <= 19.2 px -> <=22 rows incl +1)
#define TW      22         // tile cols
#define TILE_ELEMS (CHUNK * TH * TW)   // 7744 floats
#define TILE_BYTES (TILE_ELEMS * 4)    // 30976 bytes, x2 buffers = 61952 <= 64KB

struct TapParam {
    int   r0, r1, r2, r3;   // LDS row indices (4 row taps)
    int   cA, cB;           // LDS col bases (col taps cA,cA+1,cB,cB+1)
    float a0, a1, a2, a3;   // row weights (0.5 * bilinear * validity)
    float b0, b1, b2, b3;   // col weights (0.5 * bilinear * validity)
};

// Matches reference: hs = min(floor(t),62); frac = t - hs; idx = clip(hs,0,62);
// valid = (0 <= t < 64). For valid samples idx == floor(t) so frac is exact.
__device__ __forceinline__ void sample_axis(float origin, float binsz, int i,
                                            int& idx, float& frac, float& valid) {
    float t  = origin + (float)i * binsz;
    float fs = fminf(floorf(t), 62.0f);
    frac  = t - fs;
    idx   = (int)fmaxf(fs, 0.0f);
    valid = (t >= 0.0f && t < 64.0f) ? 1.0f : 0.0f;
}

__device__ __forceinline__ TapParam make_tap(int p, float y1, float x1,
                                             float bin_h, float bin_w,
                                             int h0c, int w0c) {
    const int oh = p / OHW, ow = p % OHW;
    int hiA, hiB, wiA, wiB;
    float hrA, hrB, wrA, wrB, vhA, vhB, vwA, vwB;
    sample_axis(y1, bin_h, oh,     hiA, hrA, vhA);
    sample_axis(y1, bin_h, oh + 1, hiB, hrB, vhB);
    sample_axis(x1, bin_w, ow,     wiA, wrA, vwA);
    sample_axis(x1, bin_w, ow + 1, wiB, wrB, vwB);

    TapParam tp;
    tp.r0 = min(max(hiA - h0c, 0), TH - 1);
    tp.r1 = min(tp.r0 + 1, TH - 1);
    tp.r2 = min(max(hiB - h0c, 0), TH - 1);
    tp.r3 = min(tp.r2 + 1, TH - 1);
    tp.cA = min(max(wiA - w0c, 0), TW - 2);
    tp.cB = min(max(wiB - w0c, 0), TW - 2);
    // 0.5*0.5 per (row,col) tap pair = the 0.25 of the 2x2 average
    tp.a0 = 0.5f * (1.0f - hrA) * vhA;  tp.a1 = 0.5f * hrA * vhA;
    tp.a2 = 0.5f * (1.0f - hrB) * vhB;  tp.a3 = 0.5f * hrB * vhB;
    tp.b0 = 0.5f * (1.0f - wrA) * vwA;  tp.b1 = 0.5f * wrA * vwA;
    tp.b2 = 0.5f * (1.0f - wrB) * vwB;  tp.b3 = 0.5f * wrB * vwB;
    return tp;
}

__device__ __forceinline__ float eval_tap(const float* T, const TapParam& tp) {
    const float* q0 = T + tp.r0 * TW;
    const float* q1 = T + tp.r1 * TW;
    const float* q2 = T + tp.r2 * TW;
    const float* q3 = T + tp.r3 * TW;
    float s0 = tp.b0 * q0[tp.cA] + tp.b1 * q0[tp.cA + 1] + tp.b2 * q0[tp.cB] + tp.b3 * q0[tp.cB + 1];
    float s1 = tp.b0 * q1[tp.cA] + tp.b1 * q1[tp.cA + 1] + tp.b2 * q1[tp.cB] + tp.b3 * q1[tp.cB + 1];
    float s2 = tp.b0 * q2[tp.cA] + tp.b1 * q2[tp.cA + 1] + tp.b2 * q2[tp.cB] + tp.b3 * q2[tp.cB + 1];
    float s3 = tp.b0 * q3[tp.cA] + tp.b1 * q3[tp.cA + 1] + tp.b2 * q3[tp.cB] + tp.b3 * q3[tp.cB + 1];
    return fmaf(tp.a0, s0, fmaf(tp.a1, s1, fmaf(tp.a2, s2, tp.a3 * s3)));
}

__global__ __launch_bounds__(256) void roialign_tdm_kernel(
    const float* __restrict__ feat,   // (2,512,64,64)
    const float* __restrict__ rois,   // (2000,5)
    float* __restrict__ out)          // (2000,512,7,7)
{
    // DYNAMIC shared memory: external symbol => compiler cannot prove it is
    // never written, so the ds_loads below survive (the only writer is the
    // TDM engine via the inline asm). With no static LDS in this kernel the
    // dynamic segment starts at LDS offset 0, which the TDM lds_addr uses.
    extern __shared__ float tile[];   // 2 * TILE_ELEMS floats

    const int kk   = blockIdx.x;
    const int tid  = threadIdx.x;
    const int wave = tid >> 5;
    const int lane = tid & 31;

    const float* r = rois + (size_t)kk * 5;
    const int   b  = (int)r[0];
    const float x1 = r[1] * SCALE_F, y1 = r[2] * SCALE_F;
    const float x2 = r[3] * SCALE_F, y2 = r[4] * SCALE_F;
    const float bin_h = fmaxf(y2 - y1, 0.0f) * (1.0f / 7.0f);
    const float bin_w = fmaxf(x2 - x1, 0.0f) * (1.0f / 7.0f);

    // Tile origin = first sampled row/col, clamped so tile stays inside [0,64).
    int hi0, wi0; float dumf, dumv;
    sample_axis(y1, bin_h, 0, hi0, dumf, dumv);
    sample_axis(x1, bin_w, 0, wi0, dumf, dumv);
    const int h0c = min(hi0, HF - TH);   // <= 42
    const int w0c = min(wi0, WF - TW);

    // Per-lane output pixels: p0 = lane (always < 49), p1 = lane+32 (lanes 0..16).
    const TapParam t0 = make_tap(lane, y1, x1, bin_h, bin_w, h0c, w0c);
    const TapParam t1 = make_tap(min(lane + 32, 48), y1, x1, bin_h, bin_w, h0c, w0c);
    const bool has1 = (lane + 32) < (OHW * OHW);

    const uint64_t featA = (uint64_t)(uintptr_t)feat;

    // ---- TDM descriptor issue (wave0 only). D# per CDNA5 ISA ch.8 ----
    auto issue_tdm = [&](int chunk, int buf) {
        const uint64_t ga = featA +
            4ull * ((((uint64_t)(u32)b * CCH + (u32)(chunk * CHUNK)) * HF + (u32)h0c) * WF + (u32)w0c);
        const u32 lo = (u32)ga, hi = (u32)(ga >> 32);
        v4u g0;
        g0[0] = 1u;                                   // count=1, user mode
        g0[1] = (u32)(buf * TILE_BYTES);              // lds_addr (bytes)
        g0[2] = (u32)__builtin_amdgcn_readfirstlane((int)lo);
        g0[3] = (u32)__builtin_amdgcn_readfirstlane(
                    (int)((hi & 0x01FFFFFFu) | 0x80000000u)); // ga[56:32] | type=2
        // group1: mask=0, data_size=2(4B); tensor_dim0=64, tensor_dim1=64;
        // tile_dim0=TW, tile_dim1=TH, tile_dim2=CHUNK;
        // dim0_stride=64 (row), dim1_stride=4096 (channel)
        const v8u g1 = { 0x00020000u,
                         (u32)HF << 16,               // tensor_dim0 lo16 in [31:16]
                         (u32)HF << 16,               // dim0 hi16=0 | tensor_dim1 lo16
                         (u32)TW << 16,               // dim1 hi16=0 | tile_dim0
                         ((u32)CHUNK << 16) | (u32)TH,// tile_dim1 | tile_dim2
                         (u32)WF,                     // tensor_dim0_stride lo32 = 64
                         0x1000u << 16,               // stride0 hi16=0 | stride1 lo16 (4096)
                         0u };                        // stride1 hi32 = 0
        const v4u g2 = { (u32)CCH, 0u, 0u, 0u };      // tensor_dim2=512, rest unused
        const v4u g3 = { 0u, 0u, 0u, 0u };
        asm volatile("tensor_load_to_lds %0, %1, %2, %3"
                     :: "s"(g0), "s"(g1), "s"(g2), "s"(g3)
                     : "memory");
    };

    if (tid < 32) issue_tdm(0, 0);   // prologue: chunk 0 -> buffer 0

    for (int chunk = 0; chunk < NCHUNK; ++chunk) {
        const int buf = chunk & 1;
        __syncthreads();             // buf^1 free: compute on it finished last iter
        if (tid < 32) {
            if (chunk + 1 < NCHUNK) {
                issue_tdm(chunk + 1, buf ^ 1);
                __builtin_amdgcn_s_wait_tensorcnt(1);  // chunk done (in-order), chunk+1 in flight
            } else {
                __builtin_amdgcn_s_wait_tensorcnt(0);
            }
        }
        __syncthreads();             // release compute: tile[buf] is valid

        const float* Tbase = tile + buf * TILE_ELEMS;
#pragma unroll
        for (int s = 0; s < 2; ++s) {
            const int c_local = wave + 8 * s;          // 0..15 across 8 waves
            const int c       = chunk * CHUNK + c_local;
            const float* T    = Tbase + c_local * (TH * TW);
            float* op = out + ((size_t)kk * CCH + (size_t)c) * (OHW * OHW);
            op[lane] = eval_tap(T, t0);
            if (has1) op[lane + 32] = eval_tap(T, t1);
        }
    }
}

extern "C" void kernel_launch(void* const* d_in, const int* in_sizes, int n_in,
                              void* d_out, int out_size, void* d_ws, size_t ws_size,
                              hipStream_t stream) {
    (void)in_sizes; (void)n_in; (void)out_size; (void)d_ws; (void)ws_size;
    const float* feat = (const float*)d_in[0];
    const float* rois = (const float*)d_in[1];
    float*       out  = (float*)d_out;
    dim3 grid(KROIS), block(256);
    hipLaunchKernelGGL(roialign_tdm_kernel, grid, block,
                       2 * TILE_BYTES /* dynamic LDS: 61952 B */, stream,
                       feat, rois, out);
}